// GCNEncoder_60679297958295
// MI455X (gfx1250) — compile-verified
//
#include <hip/hip_runtime.h>
#include <hip/hip_bf16.h>

#define HID 128

typedef __attribute__((ext_vector_type(2))) float v2f;
typedef __attribute__((ext_vector_type(8))) float v8f;

// ---------------------------------------------------------------------------
// Utility: fill a float buffer with a constant
// ---------------------------------------------------------------------------
__global__ void fill_f32(float* __restrict__ p, float v, int n) {
    int i = blockIdx.x * blockDim.x + threadIdx.x;
    if (i < n) p[i] = v;
}

// Pad x[N x 23] -> xp[N x 24] (column 23 zeroed) so the GEMM K-loop is exact.
__global__ void pad_x(const float* __restrict__ x, float* __restrict__ xp, int N) {
    int i = blockIdx.x * blockDim.x + threadIdx.x;   // over N*24
    if (i >= N * 24) return;
    int r = i / 24, c = i - r * 24;
    xp[i] = (c < 23) ? x[(size_t)r * 23 + c] : 0.0f;
}

// ---------------------------------------------------------------------------
// Degree accumulation: deg[dst[e]] += 1  (deg pre-filled with 1.0 = self loop)
// ---------------------------------------------------------------------------
__global__ void deg_kernel(const int* __restrict__ dst, float* __restrict__ deg, int E) {
    int e = blockIdx.x * blockDim.x + threadIdx.x;
    if (e < E) atomicAdd(&deg[dst[e]], 1.0f);
}

// deg -> dis = rsqrt(deg)   (deg >= 1 always, so matches reference's where())
__global__ void rsqrt_kernel(float* __restrict__ d, int n) {
    int i = blockIdx.x * blockDim.x + threadIdx.x;
    if (i < n) d[i] = rsqrtf(d[i]);
}

// norm[e] = dis[src[e]] * dis[dst[e]]
__global__ void norm_kernel(const int* __restrict__ src, const int* __restrict__ dst,
                            const float* __restrict__ dis, float* __restrict__ norm, int E) {
    int e = blockIdx.x * blockDim.x + threadIdx.x;
    if (e < E) norm[e] = dis[src[e]] * dis[dst[e]];
}

// ---------------------------------------------------------------------------
// WMMA fp32 GEMM:  H[M x 128] = A[M x K] * W[K x 128],  K known at compile time
// (24 or 128, multiple of 4). Block = 256 threads = 8 waves. Wave w owns the
// 16-wide column strip n0 = w*16; it loads its K/4 B-fragments ONCE into
// registers, then sweeps 4 M-tiles (64 rows per block): the hot loop is
// exactly global_load_b64 (A frag) + v_wmma_f32_16x16x4_f32, fully unrolled.
// Store epilogue: wave-uniform tile guard -> common path is one clause of
// 8 unconditional global_store_b32 (only the single tail block takes the
// per-row guarded path; 100000 % 64 = 32).
//
// fp32 WMMA VGPR layouts (ISA 7.12.2):
//   A 16x4 : lanes 0-15 -> M=lane, vgpr{0,1}=K{0,1}; lanes 16-31 -> K{2,3}
//   B 4x16 : lanes 0-15 -> N=lane, vgpr{0,1}=K{0,1}; lanes 16-31 -> K{2,3}
//   C/D    : lanes 0-15 -> N=lane, vgpr v -> M=v; lanes 16-31 -> M=8+v
// ---------------------------------------------------------------------------
template <int K>
__global__ void gemm_wmma_f32(const float* __restrict__ A, const float* __restrict__ W,
                              float* __restrict__ H, int M) {
    constexpr int KF = K / 4;                 // #WMMA steps / B fragments
    const int wave  = threadIdx.x >> 5;       // 0..7 -> column tile
    const int lane  = threadIdx.x & 31;
    const int l15   = lane & 15;
    const int lhalf = lane >> 4;              // 0: K pair {0,1}, 1: {2,3}
    const int ncol  = wave * 16 + l15;

    // ---- load B fragments once (register-resident for all 4 M tiles) ----
    v2f bf[KF];
#pragma unroll
    for (int i = 0; i < KF; ++i) {
        const int ka = i * 4 + lhalf * 2;
        bf[i].x = W[(size_t)(ka    ) * HID + ncol];
        bf[i].y = W[(size_t)(ka + 1) * HID + ncol];
    }

    const int mbase = blockIdx.x * 64;
#pragma unroll
    for (int t = 0; t < 4; ++t) {
        const int m0  = mbase + t * 16;
        const int row = m0 + l15;
        const int rc  = (row < M) ? row : (M - 1);      // clamp: branch-free load
        const float* arow = A + (size_t)rc * K + lhalf * 2;

        v8f acc = {};
#pragma unroll
        for (int i = 0; i < KF; ++i) {
            const v2f a = *(const v2f*)(arow + i * 4);  // one global_load_b64
            acc = __builtin_amdgcn_wmma_f32_16x16x4_f32(
                      /*neg_a=*/false, a, /*neg_b=*/false, bf[i],
                      /*c_mod=*/(short)0, acc, /*reuse_a=*/false, /*reuse_b=*/false);
        }

        // D-tile store: one scalar branch, unconditional clause on common path
        float* hbase = H + (size_t)(m0 + lhalf * 8) * HID + ncol;
        if (m0 + 16 <= M) {
#pragma unroll
            for (int v = 0; v < 8; ++v)
                hbase[(size_t)v * HID] = acc[v];
        } else {
#pragma unroll
            for (int v = 0; v < 8; ++v) {
                const int r = m0 + v + lhalf * 8;
                if (r < M) hbase[(size_t)v * HID] = acc[v];
            }
        }
    }
}

// ---------------------------------------------------------------------------
// Edge scatter: one wave per edge. Lane l handles features 4l..4l+3 (float4),
// so the wave reads a full coalesced 512B row of H[src] and atomically adds
// into agg[dst]. Both arrays are L2-resident (51MB each vs 192MB L2).
// ---------------------------------------------------------------------------
__global__ void scatter_edges(const float* __restrict__ H,
                              const int* __restrict__ src, const int* __restrict__ dst,
                              const float* __restrict__ norm,
                              float* __restrict__ agg, int E) {
    const int wid  = (int)((blockIdx.x * (size_t)blockDim.x + threadIdx.x) >> 5);
    const int lane = threadIdx.x & 31;
    if (wid >= E) return;
    const int   s  = src[wid];
    const int   d  = dst[wid];
    const float nv = norm[wid];
    const float4 hv = ((const float4*)(H + (size_t)s * HID))[lane];
    float* orow = agg + (size_t)d * HID + lane * 4;
    atomicAdd(orow + 0, hv.x * nv);
    atomicAdd(orow + 1, hv.y * nv);
    atomicAdd(orow + 2, hv.z * nv);
    atomicAdd(orow + 3, hv.w * nv);
}

// ---------------------------------------------------------------------------
// Finalize: out = [relu]( agg + H*dis^2 (self loop) + bias )
// ---------------------------------------------------------------------------
__global__ void finalize(const float* __restrict__ agg, const float* __restrict__ H,
                         const float* __restrict__ dis, const float* __restrict__ bias,
                         float* __restrict__ out, int total, int do_relu) {
    int idx = blockIdx.x * blockDim.x + threadIdx.x;
    if (idx >= total) return;
    const int i = idx >> 7;          // node
    const int f = idx & (HID - 1);   // feature
    const float ds = dis[i];
    float v = agg[idx] + H[idx] * ds * ds + bias[f];
    out[idx] = do_relu ? fmaxf(v, 0.0f) : v;
}

// ---------------------------------------------------------------------------
// Host launcher
// ---------------------------------------------------------------------------
extern "C" void kernel_launch(void* const* d_in, const int* in_sizes, int n_in,
                              void* d_out, int out_size, void* d_ws, size_t ws_size,
                              hipStream_t stream) {
    const float* x   = (const float*)d_in[0];
    const int*   ei  = (const int*)d_in[1];
    const float* W1  = (const float*)d_in[2]; const float* b1 = (const float*)d_in[3];
    const float* W2  = (const float*)d_in[4]; const float* b2 = (const float*)d_in[5];
    const float* W3  = (const float*)d_in[6]; const float* b3 = (const float*)d_in[7];
    const float* W4  = (const float*)d_in[8]; const float* b4 = (const float*)d_in[9];

    const int IN_DIM = 23;
    const int N = in_sizes[0] / IN_DIM;      // 100000
    const int E = in_sizes[1] / 2;           // 3200000
    const int* srcp = ei;
    const int* dstp = ei + E;

    const size_t NH = (size_t)N * HID;       // 12.8M floats

    float* h_buf = (float*)d_ws;             //  51.2 MB
    float* agg   = h_buf + NH;               //  51.2 MB
    float* act   = agg + NH;                 //  51.2 MB
    float* normv = act + NH;                 //  12.8 MB
    float* dis   = normv + E;                //   0.4 MB
    float* xpad  = dis + N;                  //   9.6 MB  (N x 24, zero-padded x)

    const int BT = 256;
    const int total = N * HID;
    const int gTot  = (total + BT - 1) / BT;
    const int gN    = (N + BT - 1) / BT;
    const int gPad  = (N * 24 + BT - 1) / BT;
    const int gE    = (E + BT - 1) / BT;
    const int gEw   = (int)(((size_t)E * 32 + BT - 1) / BT);  // one wave per edge
    const int gGemm = (N + 63) / 64;                          // 64 rows per block

    // --- normalization: deg (incl. self loop) -> dis -> edge norms ---
    fill_f32<<<gN, BT, 0, stream>>>(dis, 1.0f, N);
    deg_kernel<<<gE, BT, 0, stream>>>(dstp, dis, E);
    rsqrt_kernel<<<gN, BT, 0, stream>>>(dis, N);
    norm_kernel<<<gE, BT, 0, stream>>>(srcp, dstp, dis, normv, E);

    // --- pad input features 23 -> 24 ---
    pad_x<<<gPad, BT, 0, stream>>>(x, xpad, N);

    // --- layer 1: xpad[N x 24] ---
    gemm_wmma_f32<24><<<gGemm, BT, 0, stream>>>(xpad, W1, h_buf, N);
    fill_f32<<<gTot, BT, 0, stream>>>(agg, 0.0f, total);
    scatter_edges<<<gEw, BT, 0, stream>>>(h_buf, srcp, dstp, normv, agg, E);
    finalize<<<gTot, BT, 0, stream>>>(agg, h_buf, dis, b1, act, total, 1);

    // --- layer 2 ---
    gemm_wmma_f32<128><<<gGemm, BT, 0, stream>>>(act, W2, h_buf, N);
    fill_f32<<<gTot, BT, 0, stream>>>(agg, 0.0f, total);
    scatter_edges<<<gEw, BT, 0, stream>>>(h_buf, srcp, dstp, normv, agg, E);
    finalize<<<gTot, BT, 0, stream>>>(agg, h_buf, dis, b2, act, total, 1);

    // --- layer 3 ---
    gemm_wmma_f32<128><<<gGemm, BT, 0, stream>>>(act, W3, h_buf, N);
    fill_f32<<<gTot, BT, 0, stream>>>(agg, 0.0f, total);
    scatter_edges<<<gEw, BT, 0, stream>>>(h_buf, srcp, dstp, normv, agg, E);
    finalize<<<gTot, BT, 0, stream>>>(agg, h_buf, dis, b3, act, total, 1);

    // --- layer 4 (no relu, writes d_out) ---
    gemm_wmma_f32<128><<<gGemm, BT, 0, stream>>>(act, W4, h_buf, N);
    fill_f32<<<gTot, BT, 0, stream>>>(agg, 0.0f, total);
    scatter_edges<<<gEw, BT, 0, stream>>>(h_buf, srcp, dstp, normv, agg, E);
    finalize<<<gTot, BT, 0, stream>>>(agg, h_buf, dis, b4, (float*)d_out, total, 0);
}